// TurboDecoder_61727269978265
// MI455X (gfx1250) — compile-verified
//
#include <hip/hip_runtime.h>
#include <cstdint>
#include <cstddef>

// ---------------- problem constants ----------------
constexpr int Bn = 256;      // batch
constexpr int Kn = 6144;     // trellis length
constexpr int NUM_ITER = 6;
constexpr float NEGV = -1e30f;

// scan streaming tile (rows of [K,B] staged to LDS via TDM)
constexpr int TILE = 8;                    // rows per tile
constexpr int NT   = Kn / TILE;            // 768 tiles
constexpr int TEL  = TILE * Bn;            // elements per array per tile
// double-buffered: 2 bufs * 3 arrays * TILE*B floats = 48 KB dynamic LDS

// ---------------- log-sum-exp helpers (exact log-MAP) ----------------
__device__ __forceinline__ float lse2(float x, float y) {
    float m = fmaxf(x, y);
    float d = fabsf(x - y);
    return m + __logf(1.0f + __expf(-d));
}

__device__ __forceinline__ float lse8(float x0, float x1, float x2, float x3,
                                      float x4, float x5, float x6, float x7) {
    float m = fmaxf(fmaxf(fmaxf(x0, x1), fmaxf(x2, x3)),
                    fmaxf(fmaxf(x4, x5), fmaxf(x6, x7)));
    float s = __expf(x0 - m) + __expf(x1 - m) + __expf(x2 - m) + __expf(x3 - m) +
              __expf(x4 - m) + __expf(x5 - m) + __expf(x6 - m) + __expf(x7 - m);
    return m + __logf(s);
}

// ---------------- CDNA5 Tensor Data Mover: 1-D async copy global -> LDS ----
// D# layout per CDNA5 ISA ch.7/8 (async_tensor): group0 = {flags, lds_addr,
// global_addr[31:0], global_addr[56:32] | type=2<<30}; group1 carries
// data_size / tensor_dim0 / tile_dim0. A contiguous row of `tile_elems`
// fp32 values is described as a 1-D tile (tile_dim1/2 = 0).
typedef unsigned int v4u __attribute__((ext_vector_type(4)));
typedef int          v8i __attribute__((ext_vector_type(8)));
typedef int          v4i __attribute__((ext_vector_type(4)));

__device__ __forceinline__ void tdm_load_1d(const void* gsrc, const void* ldsdst,
                                            unsigned tile_elems, unsigned tensor_elems) {
    unsigned long long ga = (unsigned long long)(uintptr_t)gsrc;
    v4u g0;
    g0.x = 1u;                                       // count=1 valid descriptor
    g0.y = (unsigned)(uintptr_t)ldsdst;              // low 32 bits of flat LDS ptr == LDS byte addr
    g0.z = (unsigned)ga;                             // global_addr[31:0]
    g0.w = (unsigned)((ga >> 32) & 0x01FFFFFFu) | 0x80000000u; // [56:32] | type=2
    v8i g1;
    g1[0] = 2 << 16;                                 // data_size = 4 bytes
    g1[1] = (int)((tensor_elems & 0xFFFFu) << 16);   // tensor_dim0[15:0]  @ bits 63:48
    g1[2] = (int)((tensor_elems >> 16) & 0xFFFFu)    // tensor_dim0[31:16] @ bits 79:64
          | (1 << 16);                               // tensor_dim1 = 1    @ bits 111:80 (low half)
    g1[3] = (int)((tile_elems & 0xFFFFu) << 16);     // tile_dim0          @ bits 127:112
    g1[4] = 0;                                       // tile_dim1 = tile_dim2 = 0 (1-D)
    g1[5] = (int)tensor_elems;                       // tensor_dim0_stride (unused, 1 row)
    g1[6] = 0;
    g1[7] = 0;
    v4i z4 = {0, 0, 0, 0};
    v8i z8 = {0, 0, 0, 0, 0, 0, 0, 0};
    // clang-23 / therock-10.0 lane: 6-arg form
    // (uint32x4 g0, int32x8 g1, int32x4 g2, int32x4 g3, int32x8, i32 cpol)
    __builtin_amdgcn_tensor_load_to_lds(g0, g1, z4, z4, z8, 0);
}

// ---------------- prep: negate, transpose to [K,B], interleave, inv perm ---
__global__ void __launch_bounds__(Bn) prep_kernel(
    const float* __restrict__ in, const int* __restrict__ perm,
    float* __restrict__ ls1, float* __restrict__ lp1, float* __restrict__ lp2,
    float* __restrict__ ls2, float* __restrict__ la1, int* __restrict__ inv)
{
    const int k = blockIdx.x;
    const int b = threadIdx.x;
    const float* inb = in + (size_t)b * (3 * Kn);
    const size_t o = (size_t)k * Bn + b;
    ls1[o] = -inb[3 * k + 0];
    lp1[o] = -inb[3 * k + 1];
    lp2[o] = -inb[3 * k + 2];
    const int pk = perm[k];
    ls2[o] = -inb[3 * pk + 0];
    la1[o] = 0.0f;                 // initial a-priori (le2_de = 0)
    if (b == 0) inv[pk] = k;       // inverse interleaver
}

// ---------------- BCJR scans (fwd alpha / bwd beta), TDM-staged -----------
// Trellis (MU=3, G0=1011, G1=1101), branch metrics: g00 = (lsu+lp)/2,
// g01 = (lsu-lp)/2, and g(u=1,*) = -g(u=0,*).
template <bool FWD>
__device__ __forceinline__ void scan_dir(const float* __restrict__ ls,
                                         const float* __restrict__ lp,
                                         const float* __restrict__ la,
                                         float* __restrict__ traj,  // [K][B][8]
                                         float* sm)
{
    const int b = threadIdx.x;
    const bool issuer = (threadIdx.x < 32);   // wave 0 drives the TDM

    auto base_row = [](int t) { return FWD ? t * TILE : Kn - (t + 1) * TILE; };

    auto issue_tile = [&](int t, int buf) {
        const int r = base_row(t);
        const unsigned rem = (unsigned)((Kn - r) * Bn);
        tdm_load_1d(ls + (size_t)r * Bn, sm + (buf * 3 + 0) * TEL, (unsigned)TEL, rem);
        tdm_load_1d(lp + (size_t)r * Bn, sm + (buf * 3 + 1) * TEL, (unsigned)TEL, rem);
        tdm_load_1d(la + (size_t)r * Bn, sm + (buf * 3 + 2) * TEL, (unsigned)TEL, rem);
    };

    if (issuer) issue_tile(0, 0);

    float a0, a1, a2, a3, a4, a5, a6, a7;
    if (FWD) { a0 = 0.0f; a1 = a2 = a3 = a4 = a5 = a6 = a7 = NEGV; }
    else     { a0 = a1 = a2 = a3 = a4 = a5 = a6 = a7 = 0.0f; }

    for (int t = 0; t < NT; ++t) {
        const int cur = t & 1;
        if (issuer) {
            if (t + 1 < NT) {
                issue_tile(t + 1, 1 - cur);
                __builtin_amdgcn_s_wait_tensorcnt(3);   // tile t's 3 loads done
            } else {
                __builtin_amdgcn_s_wait_tensorcnt(0);
            }
        }
        __syncthreads();

        const float* sls = sm + (cur * 3 + 0) * TEL;
        const float* slp = sm + (cur * 3 + 1) * TEL;
        const float* sla = sm + (cur * 3 + 2) * TEL;
        const int r0 = base_row(t);

#pragma unroll
        for (int i = 0; i < TILE; ++i) {
            const int row = FWD ? i : (TILE - 1 - i);
            const int k = r0 + row;
            const float lsv = sls[row * Bn + b];
            const float lpv = slp[row * Bn + b];
            const float lav = sla[row * Bn + b];
            const float g00 = 0.5f * (lsv + lav + lpv);
            const float g01 = 0.5f * (lsv + lav - lpv);

            // store the state entering step k (matches jax.lax.scan outputs)
            float4* dst = reinterpret_cast<float4*>(traj) + ((size_t)k * Bn + b) * 2;
            float4 v0; v0.x = a0; v0.y = a1; v0.z = a2; v0.w = a3;
            float4 v1; v1.x = a4; v1.y = a5; v1.z = a6; v1.w = a7;
            dst[0] = v0; dst[1] = v1;

            float n0, n1, n2, n3, n4, n5, n6, n7;
            if (FWD) {
                n0 = lse2(a0 + g00, a1 - g00);
                n1 = lse2(a2 - g01, a3 + g01);
                n2 = lse2(a4 + g01, a5 - g01);
                n3 = lse2(a6 - g00, a7 + g00);
                n4 = lse2(a0 - g00, a1 + g00);
                n5 = lse2(a2 + g01, a3 - g01);
                n6 = lse2(a4 - g01, a5 + g01);
                n7 = lse2(a6 + g00, a7 - g00);
            } else {
                n0 = lse2(a0 + g00, a4 - g00);
                n1 = lse2(a4 + g00, a0 - g00);
                n2 = lse2(a5 + g01, a1 - g01);
                n3 = lse2(a1 + g01, a5 - g01);
                n4 = lse2(a2 + g01, a6 - g01);
                n5 = lse2(a6 + g01, a2 - g01);
                n6 = lse2(a7 + g00, a3 - g00);
                n7 = lse2(a3 + g00, a7 - g00);
            }
            const float m = fmaxf(fmaxf(fmaxf(n0, n1), fmaxf(n2, n3)),
                                  fmaxf(fmaxf(n4, n5), fmaxf(n6, n7)));
            a0 = n0 - m; a1 = n1 - m; a2 = n2 - m; a3 = n3 - m;
            a4 = n4 - m; a5 = n5 - m; a6 = n6 - m; a7 = n7 - m;
        }
        __syncthreads();
    }
}

__global__ void __launch_bounds__(Bn) bcjr_scan_kernel(
    const float* __restrict__ ls, const float* __restrict__ lp,
    const float* __restrict__ la, float* __restrict__ alphas,
    float* __restrict__ betas)
{
    extern __shared__ float sm[];
    if (blockIdx.x == 0) scan_dir<true>(ls, lp, la, alphas, sm);
    else                 scan_dir<false>(ls, lp, la, betas, sm);
}

// ---------------- combine: posterior LLR + permuted extrinsic scatter -----
__global__ void __launch_bounds__(Bn) bcjr_combine_kernel(
    const float* __restrict__ ls, const float* __restrict__ lp,
    const float* __restrict__ la, const float* __restrict__ alphas,
    const float* __restrict__ betas, const int* __restrict__ map,
    float* __restrict__ extr, float* __restrict__ final_out)
{
    const int k = blockIdx.x;
    const int b = threadIdx.x;
    const size_t idx = (size_t)k * Bn + b;
    const float lsv = ls[idx], lpv = lp[idx], lav = la[idx];
    const float g00 = 0.5f * (lsv + lav + lpv);
    const float g01 = 0.5f * (lsv + lav - lpv);

    const float4* ap = reinterpret_cast<const float4*>(alphas) + idx * 2;
    const float4* bp = reinterpret_cast<const float4*>(betas) + idx * 2;
    const float4 A0 = ap[0], A1 = ap[1];
    const float4 B0 = bp[0], B1 = bp[1];
    const float a0 = A0.x, a1 = A0.y, a2 = A0.z, a3 = A0.w;
    const float a4 = A1.x, a5 = A1.y, a6 = A1.z, a7 = A1.w;
    const float e0 = B0.x, e1 = B0.y, e2 = B0.z, e3 = B0.w;
    const float e4 = B1.x, e5 = B1.y, e6 = B1.z, e7 = B1.w;

    // u = 0 branch terms: alpha[s] + gamma(s,0) + beta_{k+1}[nxt(s,0)]
    const float t0 = a0 + g00 + e0, t1 = a1 + g00 + e4;
    const float t2 = a2 + g01 + e5, t3 = a3 + g01 + e1;
    const float t4 = a4 + g01 + e2, t5 = a5 + g01 + e6;
    const float t6 = a6 + g00 + e7, t7 = a7 + g00 + e3;
    // u = 1 branch terms
    const float s0 = a0 - g00 + e4, s1 = a1 - g00 + e0;
    const float s2 = a2 - g01 + e1, s3 = a3 - g01 + e5;
    const float s4 = a4 - g01 + e6, s5 = a5 - g01 + e2;
    const float s6 = a6 - g00 + e3, s7 = a7 - g00 + e7;

    const float L = lse8(t0, t1, t2, t3, t4, t5, t6, t7) -
                    lse8(s0, s1, s2, s3, s4, s5, s6, s7);

    const int mk = map[k];
    extr[(size_t)mk * Bn + b] = L - lav - lsv;      // interleaved extrinsic
    if (final_out) final_out[(size_t)b * Kn + mk] = -L;
}

// ---------------- host orchestration --------------------------------------
extern "C" void kernel_launch(void* const* d_in, const int* in_sizes, int n_in,
                              void* d_out, int out_size, void* d_ws, size_t ws_size,
                              hipStream_t stream)
{
    (void)in_sizes; (void)n_in; (void)out_size; (void)ws_size;
    const float* in  = (const float*)d_in[0];   // [B, 3K] float32
    const int*  perm = (const int*)d_in[1];     // [K] int32
    float* out = (float*)d_out;                 // [B, K] float32

    float* ws = (float*)d_ws;
    const size_t KB = (size_t)Kn * Bn;
    float* ls1 = ws;
    float* lp1 = ls1 + KB;
    float* lp2 = lp1 + KB;
    float* ls2 = lp2 + KB;
    float* la1 = ls2 + KB;
    float* la2 = la1 + KB;
    float* alphas = la2 + KB;          // [K][B][8]
    float* betas  = alphas + 8 * KB;   // [K][B][8]
    int*   inv    = (int*)(betas + 8 * KB);

    prep_kernel<<<Kn, Bn, 0, stream>>>(in, perm, ls1, lp1, lp2, ls2, la1, inv);

    const size_t smem = (size_t)2 * 3 * TILE * Bn * sizeof(float);  // 48 KB
    for (int it = 0; it < NUM_ITER; ++it) {
        // decoder 1
        bcjr_scan_kernel<<<2, Bn, smem, stream>>>(ls1, lp1, la1, alphas, betas);
        bcjr_combine_kernel<<<Kn, Bn, 0, stream>>>(ls1, lp1, la1, alphas, betas,
                                                   inv, la2, nullptr);
        // decoder 2 (writes output; last iteration's values are final)
        bcjr_scan_kernel<<<2, Bn, smem, stream>>>(ls2, lp2, la2, alphas, betas);
        bcjr_combine_kernel<<<Kn, Bn, 0, stream>>>(ls2, lp2, la2, alphas, betas,
                                                   perm, la1, out);
    }
}